// GraphConvClf_50130858279304
// MI455X (gfx1250) — compile-verified
//
#include <hip/hip_runtime.h>
#include <hip/hip_bf16.h>

typedef __attribute__((ext_vector_type(2))) float v2f;
typedef __attribute__((ext_vector_type(8))) float v8f;

#define WMMA_F32_4(A, B, C) \
    __builtin_amdgcn_wmma_f32_16x16x4_f32(false, (A), false, (B), (short)0, (C), false, false)

// ---------------------------------------------------------------------------
// Fused dual GEMM:  acc = x @ w0 + b0   and   x1 = x @ w1 + b1
// x: [V, DIN] row-major, w*: [DIN, DOUT] row-major. DIN/DOUT are template
// params (layer shapes are static), so the K-loop is branch-free and all
// store offsets are compile-time immediates.
//
// Block = 128 threads = 4 waves; a block owns 64 output columns whose w0/w1
// slices are staged once into LDS (layout lw[n][k], row stride LSTR even and
// == 4 (mod 64) dwords -> a wave's b64 reads hit all 64 banks exactly once,
// 8B-aligned). Each block loops over "pairs" of M-tiles: a wave computes TWO
// 16x16 tiles of BOTH outputs per iteration, reusing one pair of B ds-loads
// for 4 V_WMMA_F32_16X16X4_F32 ops (exact fp32).
// Epilogue: biases are folded into the accumulator VGPRs first, then a pure
// burst of immediate-offset stores (no single-temp serialization / xcnt
// stalls). Full-pair stores take a scalar-branch fast path.
// EXEC stays all-1s around WMMA.
// ---------------------------------------------------------------------------
template <int DIN, int DOUT>
__global__ __launch_bounds__(128)
void gconv_dual_gemm(const float* __restrict__ x,
                     const float* __restrict__ w0, const float* __restrict__ b0,
                     const float* __restrict__ w1, const float* __restrict__ b1,
                     float* __restrict__ acc, float* __restrict__ x1,
                     int nV, int pairsPerBlock, int numPairs)
{
    constexpr int DINP = (DIN + 3) & ~3;                 // K padded to mult of 4
    constexpr int LSTR = (DINP <= 4) ? 4 : DINP + 4;     // LDS row stride (dwords)
    __shared__ float lw0[64 * LSTR];
    __shared__ float lw1[64 * LSTR];

    const int tid = threadIdx.x;
    const int n0  = blockIdx.y * 64;

    // Stage 64-column slices of w0/w1 into LDS (zero-pad K beyond DIN).
    for (int i = tid; i < 64 * DINP; i += 128) {
        const int nn = i / DINP;
        const int k  = i % DINP;
        const bool ok = (k < DIN);
        lw0[nn * LSTR + k] = ok ? w0[k * DOUT + n0 + nn] : 0.0f;
        lw1[nn * LSTR + k] = ok ? w1[k * DOUT + n0 + nn] : 0.0f;
    }
    __syncthreads();

    const int lane = tid & 31;
    const int wave = tid >> 5;
    const int half = lane >> 4;       // 0: lanes 0-15, 1: lanes 16-31
    const int l16  = lane & 15;
    const int n    = n0 + wave * 16 + l16;       // per-lane output column
    const float bias0 = b0[n];
    const float bias1 = b1[n];
    const float* __restrict__ lb0 = lw0 + (wave * 16 + l16) * LSTR;
    const float* __restrict__ lb1 = lw1 + (wave * 16 + l16) * LSTR;

    for (int t = 0; t < pairsPerBlock; ++t) {
        const int pair = blockIdx.x * pairsPerBlock + t;  // uniform per block
        if (pair >= numPairs) break;                      // scalar branch only
        const int row0 = pair * 32;

        const int m0 = row0 + l16;                        // A rows, tile 0 / 1
        const int m1 = row0 + 16 + l16;
        const long long arow0 = (long long)((m0 < nV) ? m0 : nV - 1) * DIN;
        const long long arow1 = (long long)((m1 < nV) ? m1 : nV - 1) * DIN;

        v8f c00 = {}, c01 = {}, c10 = {}, c11 = {};
#pragma unroll
        for (int k0 = 0; k0 < DINP; k0 += 4) {
            const int kk = k0 + 2 * half;                 // K split across halves

            v2f a0, a1;
            if constexpr (DIN % 4 == 0) {
                const float2 t0 = *(const float2*)(x + arow0 + kk);
                const float2 t1 = *(const float2*)(x + arow1 + kk);
                a0.x = t0.x; a0.y = t0.y;
                a1.x = t1.x; a1.y = t1.y;
            } else {
                // Branch-free tail: clamped always-safe loads + v_cndmask selects.
                const int ix = (kk     < DIN) ? kk     : (DIN - 1);
                const int iy = (kk + 1 < DIN) ? kk + 1 : (DIN - 1);
                const float v0x = x[arow0 + ix], v0y = x[arow0 + iy];
                const float v1x = x[arow1 + ix], v1y = x[arow1 + iy];
                a0.x = (kk     < DIN) ? v0x : 0.0f;
                a0.y = (kk + 1 < DIN) ? v0y : 0.0f;
                a1.x = (kk     < DIN) ? v1x : 0.0f;
                a1.y = (kk + 1 < DIN) ? v1y : 0.0f;
            }

            // One pair of B ds-loads feeds 4 WMMAs (conflict-free banking).
            const float2 tb0 = *(const float2*)(lb0 + kk);
            const float2 tb1 = *(const float2*)(lb1 + kk);
            v2f bb0; bb0.x = tb0.x; bb0.y = tb0.y;
            v2f bb1; bb1.x = tb1.x; bb1.y = tb1.y;

            c00 = WMMA_F32_4(a0, bb0, c00);
            c01 = WMMA_F32_4(a0, bb1, c01);
            c10 = WMMA_F32_4(a1, bb0, c10);
            c11 = WMMA_F32_4(a1, bb1, c11);
        }

        // Fold biases into the accumulator registers first (distinct VGPRs),
        // so the store burst below has no shared-temp dependencies.
#pragma unroll
        for (int r = 0; r < 8; ++r) {
            c00[r] += bias0;
            c01[r] += bias1;
            c10[r] += bias0;
            c11[r] += bias1;
        }

        // C/D layout: VGPR r holds row (half*8 + r); tile 1 rows are +16.
        const long long obase = (long long)(row0 + half * 8) * DOUT + n;
        float* __restrict__ pA = acc + obase;
        float* __restrict__ pX = x1 + obase;
        if (row0 + 32 <= nV) {
            // Fast path (uniform branch): pure immediate-offset store burst.
#pragma unroll
            for (int r = 0; r < 8; ++r) {
                pA[r * DOUT]        = c00[r];
                pX[r * DOUT]        = c01[r];
                pA[(16 + r) * DOUT] = c10[r];
                pX[(16 + r) * DOUT] = c11[r];
            }
        } else {
            // Cold tail: per-row guards.
#pragma unroll
            for (int r = 0; r < 8; ++r) {
                const int ma = row0 + half * 8 + r;
                if (ma < nV) {
                    pA[r * DOUT] = c00[r];
                    pX[r * DOUT] = c01[r];
                }
                if (ma + 16 < nV) {
                    pA[(16 + r) * DOUT] = c10[r];
                    pX[(16 + r) * DOUT] = c11[r];
                }
            }
        }
    }
}

// ---------------------------------------------------------------------------
// Scatter-add over undirected edges (both directions):
//   acc[v] += x1[u]  for each directed (v,u).
// One unit of work = (directed edge, 4-channel group); dout/4 is a power of
// two so decomposition is shift/mask. float4 row read (coalesced across the
// channel-group threads) + 4 global_atomic_add_f32.
// ---------------------------------------------------------------------------
__global__ __launch_bounds__(256)
void scatter_edges(const int* __restrict__ edges,
                   const float* __restrict__ x1,
                   float* __restrict__ acc,
                   int nE2, int log4, int dout)
{
    const int mask = (1 << log4) - 1;
    const long long nWork = (long long)nE2 << log4;
    long long idx = (long long)blockIdx.x * blockDim.x + threadIdx.x;
    const long long stride = (long long)gridDim.x * blockDim.x;
    for (; idx < nWork; idx += stride) {
        const int cg  = (int)idx & mask;
        const int e2  = (int)(idx >> log4);
        const int e   = e2 >> 1;
        const int dir = e2 & 1;
        const int v = edges[2 * e + dir];
        const int u = edges[2 * e + (dir ^ 1)];
        const float4 val = *(const float4*)(x1 + (long long)u * dout + cg * 4);
        float* dst = acc + (long long)v * dout + cg * 4;
        atomicAdd(dst + 0, val.x);
        atomicAdd(dst + 1, val.y);
        atomicAdd(dst + 2, val.z);
        atomicAdd(dst + 3, val.w);
    }
}

__global__ __launch_bounds__(256)
void relu_inplace(float4* __restrict__ p, long long n4)
{
    long long idx = (long long)blockIdx.x * blockDim.x + threadIdx.x;
    const long long stride = (long long)gridDim.x * blockDim.x;
    for (; idx < n4; idx += stride) {
        float4 v = p[idx];
        v.x = fmaxf(v.x, 0.0f); v.y = fmaxf(v.y, 0.0f);
        v.z = fmaxf(v.z, 0.0f); v.w = fmaxf(v.w, 0.0f);
        p[idx] = v;
    }
}

// ---------------------------------------------------------------------------
// Segment sum/max pooling. verts_idx is sorted, so each block reduces a run of
// `vpb` consecutive vertices in registers (one channel per thread, d == 256 ==
// blockDim) and flushes atomics only on segment transitions.
// maxs uses int-bitcast atomicMax (valid: values >= 0 post-ReLU, init 0).
// ---------------------------------------------------------------------------
__global__ __launch_bounds__(256)
void pool_segments(const float* __restrict__ x, const int* __restrict__ seg,
                   float* __restrict__ sums, float* __restrict__ maxs,
                   float* __restrict__ counts, int nV, int d, int vpb)
{
    const int c  = threadIdx.x;
    const int v0 = blockIdx.x * vpb;
    if (v0 >= nV) return;
    const int vend = (v0 + vpb < nV) ? (v0 + vpb) : nV;

    int   cur = seg[v0];
    float s = 0.0f, m = 0.0f;
    int   cnt = 0;
    for (int v = v0; v < vend; ++v) {
        const int sg = seg[v];
        if (sg != cur) {
            atomicAdd(&sums[(long long)cur * d + c], s);
            atomicMax((int*)&maxs[(long long)cur * d + c], __float_as_int(m));
            if (c == 0) atomicAdd(&counts[cur], (float)cnt);
            cur = sg; s = 0.0f; m = 0.0f; cnt = 0;
        }
        const float val = x[(long long)v * d + c];
        s += val;
        m = fmaxf(m, val);
        ++cnt;
    }
    atomicAdd(&sums[(long long)cur * d + c], s);
    atomicMax((int*)&maxs[(long long)cur * d + c], __float_as_int(m));
    if (c == 0) atomicAdd(&counts[cur], (float)cnt);
}

__global__ __launch_bounds__(256)
void pool_finalize(const float* __restrict__ sums, const float* __restrict__ maxs,
                   const float* __restrict__ counts, float* __restrict__ feat,
                   int Bn, int d)
{
    const int t = blockIdx.x * blockDim.x + threadIdx.x;
    if (t >= Bn * d) return;
    const int b = t / d, c = t % d;
    const float cnt = fmaxf(counts[b], 1.0f);
    feat[(long long)b * 2 * d + c]     = sums[t] / cnt;
    feat[(long long)b * 2 * d + d + c] = maxs[t];
}

__global__ __launch_bounds__(256)
void fc_layer(const float* __restrict__ in, const float* __restrict__ w,
              const float* __restrict__ bias, float* __restrict__ out,
              int Bn, int K, int N, int do_relu)
{
    const int t = blockIdx.x * blockDim.x + threadIdx.x;
    if (t >= Bn * N) return;
    const int b = t / N, j = t % N;
    float accv = bias[j];
    for (int k = 0; k < K; ++k)
        accv += in[(long long)b * K + k] * w[(long long)k * N + j];
    out[t] = do_relu ? fmaxf(accv, 0.0f) : accv;
}

// ---------------------------------------------------------------------------

static inline long long ll_min(long long a, long long b) { return a < b ? a : b; }

static inline int ilog2i(int v) { int l = 0; while ((1 << l) < v) ++l; return l; }

extern "C" void kernel_launch(void* const* d_in, const int* in_sizes, int n_in,
                              void* d_out, int out_size, void* d_ws, size_t ws_size,
                              hipStream_t stream)
{
    const float* verts = (const float*)d_in[0];
    const int*   edges = (const int*)d_in[1];
    const int*   vidx  = (const int*)d_in[2];
    const float* w0_0 = (const float*)d_in[3];
    const float* b0_0 = (const float*)d_in[4];
    const float* w1_0 = (const float*)d_in[5];
    const float* b1_0 = (const float*)d_in[6];
    const float* w0_1 = (const float*)d_in[7];
    const float* b0_1 = (const float*)d_in[8];
    const float* w1_1 = (const float*)d_in[9];
    const float* b1_1 = (const float*)d_in[10];
    const float* w0_2 = (const float*)d_in[11];
    const float* b0_2 = (const float*)d_in[12];
    const float* w1_2 = (const float*)d_in[13];
    const float* b1_2 = (const float*)d_in[14];
    const float* fc1_w = (const float*)d_in[15];
    const float* fc1_b = (const float*)d_in[16];
    const float* fc2_w = (const float*)d_in[17];
    const float* fc2_b = (const float*)d_in[18];

    const int nV = in_sizes[0] / 3;
    const int E  = in_sizes[1] / 2;
    const int Bn = 64, DLAST = 256, FC1N = 1024, NCLS = 10;

    char* ws = (char*)d_ws;
    const size_t bufBytes = (size_t)nV * 256 * sizeof(float);
    float* buf0 = (float*)(ws);
    float* buf1 = (float*)(ws + bufBytes);       // x1 scratch
    float* buf2 = (float*)(ws + 2 * bufBytes);
    char*  smallp = ws + 3 * bufBytes;
    float* sums   = (float*)smallp;              // [64, 256]
    float* maxs   = sums + Bn * DLAST;           // [64, 256]
    float* counts = maxs + Bn * DLAST;           // [64]
    float* feat   = counts + Bn;                 // [64, 512]
    float* h1     = feat + Bn * 2 * DLAST;       // [64, 1024]

    const int numPairs = (nV + 31) / 32;         // pairs of 16-row M-tiles
    const int pairsPerBlock = 4;                 // 128 rows per block
    const int gx = (numPairs + pairsPerBlock - 1) / pairsPerBlock;

    auto post_gemm = [&](int dout, float* accb, float* x1b) {
        const int dout4 = dout / 4;
        const long long nWork = (long long)E * 2 * dout4;
        const int sblocks = (int)ll_min((nWork + 255) / 256, 1 << 20);
        scatter_edges<<<sblocks, 256, 0, stream>>>(edges, x1b, accb,
                                                   E * 2, ilog2i(dout4), dout);
        const long long n4 = (long long)nV * dout / 4;
        const int rblocks = (int)ll_min((n4 + 255) / 256, 1 << 20);
        relu_inplace<<<rblocks, 256, 0, stream>>>((float4*)accb, n4);
    };

    // Layer 0:  verts[V,3] -> buf0[V,64]
    gconv_dual_gemm<3, 64><<<dim3(gx, 1), 128, 0, stream>>>(
        verts, w0_0, b0_0, w1_0, b1_0, buf0, buf1, nV, pairsPerBlock, numPairs);
    post_gemm(64, buf0, buf1);

    // Layer 1:  buf0[V,64] -> buf2[V,128]
    gconv_dual_gemm<64, 128><<<dim3(gx, 2), 128, 0, stream>>>(
        buf0, w0_1, b0_1, w1_1, b1_1, buf2, buf1, nV, pairsPerBlock, numPairs);
    post_gemm(128, buf2, buf1);

    // Layer 2:  buf2[V,128] -> buf0[V,256]
    gconv_dual_gemm<128, 256><<<dim3(gx, 4), 128, 0, stream>>>(
        buf2, w0_2, b0_2, w1_2, b1_2, buf0, buf1, nV, pairsPerBlock, numPairs);
    post_gemm(256, buf0, buf1);

    // Pooling (zero sums/maxs/counts first; graph-capture-safe async memset).
    hipMemsetAsync(smallp, 0, (size_t)(Bn * DLAST * 2 + Bn) * sizeof(float), stream);
    const int vpb = 128;
    pool_segments<<<(nV + vpb - 1) / vpb, 256, 0, stream>>>(buf0, vidx, sums, maxs,
                                                            counts, nV, DLAST, vpb);
    pool_finalize<<<(Bn * DLAST + 255) / 256, 256, 0, stream>>>(sums, maxs, counts,
                                                                feat, Bn, DLAST);
    // FC head
    fc_layer<<<(Bn * FC1N + 255) / 256, 256, 0, stream>>>(feat, fc1_w, fc1_b, h1,
                                                          Bn, 2 * DLAST, FC1N, 1);
    fc_layer<<<(Bn * NCLS + 255) / 256, 256, 0, stream>>>(h1, fc2_w, fc2_b,
                                                          (float*)d_out,
                                                          Bn, FC1N, NCLS, 0);
}